// KLLoss_24507083391381
// MI455X (gfx1250) — compile-verified
//
#include <hip/hip_runtime.h>
#include <math.h>

// ---------------------------------------------------------------------------
// KL loss: one streaming pass per 8192x8192 fp32 matrix (512MB total reads,
// ~22us at 23.3 TB/s is the roofline floor).
//   - async global->LDS staging (GLOBAL_LOAD_ASYNC_TO_LDS_B128, double
//     buffered, S_WAIT_ASYNCCNT)
//   - branchless chunked logsumexp per row (slice max -> single rescale ->
//     one v_exp per element) reading LDS via ds_load_b128
//   - masked / weighted row sums via V_WMMA_F32_16X16X4_F32 on the same tile
// ---------------------------------------------------------------------------

#ifndef USE_ASYNC_LDS
#define USE_ASYNC_LDS 1
#endif

#define NN 8192
#define ROWS_PER_BLK 16
#define KC 128                      // columns per LDS chunk
#define NCHUNK (NN / KC)            // 64
#define NBLK_PER_MAT (NN / ROWS_PER_BLK)  // 512

typedef float v2f __attribute__((ext_vector_type(2)));
typedef float v8f __attribute__((ext_vector_type(8)));

#if defined(__has_builtin)
#if __has_builtin(__builtin_amdgcn_wmma_f32_16x16x4_f32)
#define HAVE_WMMA_F32X4 1
#endif
#endif

#define E10D   22026.465794806718      // e^10
#define E10M1F 22025.465794806718f     // e^10 - 1

// ---------------------------------------------------------------------------
// Kernel 1: per-class counts and constants (single block, deterministic).
// ---------------------------------------------------------------------------
__global__ __launch_bounds__(256)
void kl_class_stats(const int* __restrict__ labels,
                    float* __restrict__ gcl, float* __restrict__ fcl,
                    float* __restrict__ tcl, double* __restrict__ scal)
{
    __shared__ int    cnt[128];
    __shared__ double eC[128];
    __shared__ double gC[128];
    __shared__ double Gsh;
    const int t = threadIdx.x;
    if (t < 128) cnt[t] = 0;
    __syncthreads();
    for (int k = t; k < NN; k += 256) atomicAdd(&cnt[labels[k] & 127], 1);
    __syncthreads();
    if (t < 128) {
        const int c = cnt[t];
        const double g = 1.0 / ((double)c * E10D + (double)(NN - c));
        const double f = E10D * g;
        gcl[t] = (float)g;
        fcl[t] = (float)f;
        eC[t]  = (c > 0) ? (double)c * ((double)c * f * log(f) +
                                        (double)(NN - c) * g * log(g))
                         : 0.0;
        gC[t]  = (double)c * g;
    }
    __syncthreads();
    if (t == 0) {
        double E = 0.0, G = 0.0;
        for (int i = 0; i < 128; ++i) { E += eC[i]; G += gC[i]; }
        scal[0] = E;
        Gsh = G;
    }
    __syncthreads();
    if (t < 128) {
        const int c = cnt[t];
        const double g = 1.0 / ((double)c * E10D + (double)(NN - c));
        tcl[t] = (float)(Gsh + (E10D - 1.0) * (double)c * g);
    }
}

// ---------------------------------------------------------------------------
// Kernel 2: main streaming pass. 1024 blocks x 256 threads.
// ---------------------------------------------------------------------------
__global__ __launch_bounds__(256)
void kl_main(const float* __restrict__ p1, const float* __restrict__ p2,
             const int* __restrict__ labels,
             const float* __restrict__ gcl, const float* __restrict__ fcl,
             const float* __restrict__ tcl,
             float* __restrict__ blockOut)
{
    __shared__ __attribute__((aligned(16))) float buf[2][ROWS_PER_BLK * KC];
    __shared__ int   shLab[NN];
    __shared__ float shGcl[128];
    __shared__ float shPartS[8 * 16];
    __shared__ float shPartM[8 * 16];
    __shared__ float shRowL[16];
    __shared__ float shContrib[16];

    const int tid = threadIdx.x;
    const int bid = blockIdx.x;
    const int mat = bid >> 9;                               // 0: pred1, 1: pred2
    const int rowBase = (bid & (NBLK_PER_MAT - 1)) * ROWS_PER_BLK;
    const float* __restrict__ src = mat ? p2 : p1;

    for (int k = tid; k < NN; k += 256) shLab[k] = labels[k] & 127;
    if (tid < 128) { shGcl[tid] = gcl[tid]; shPartS[tid] = 0.f; shPartM[tid] = 0.f; }
    __syncthreads();

    const int lane = tid & 31;
    const int wave = tid >> 5;
    const int nIdx = lane & 15;    // WMMA: M row for A, N col for B/D
    const int kh   = lane >> 4;    // WMMA: K-half selector
    const int labN = shLab[rowBase + nIdx];   // label of output column n
    const int rRow = tid >> 4;                // row for the logsumexp slice
    const int labR = shLab[rowBase + rRow];
    (void)labN; (void)labR; (void)kh; (void)nIdx;

    float m = -INFINITY, s = 0.f;
#if HAVE_WMMA_F32X4
    v8f accM = {0.f, 0.f, 0.f, 0.f, 0.f, 0.f, 0.f, 0.f};
    v8f accS = accM;
#else
    float pS = 0.f, pM = 0.f;
#endif

    auto issueChunk = [&](int c, int b) {
        const float* base = src + (size_t)rowBase * NN + (size_t)c * KC;
#if USE_ASYNC_LDS
#pragma unroll
        for (int j = 0; j < 2; ++j) {
            const int flat = j * 256 + tid;        // 512 x 16B = 8KB tile
            const int r    = flat >> 5;
            const int col  = (flat & 31) * 4;
            const unsigned ldsAddr =
                (unsigned)(size_t)&buf[b][r * KC + col];
            const unsigned long long ga =
                (unsigned long long)(size_t)(base + (size_t)r * NN + col);
            asm volatile("global_load_async_to_lds_b128 %0, %1, off"
                         :: "v"(ldsAddr), "v"(ga) : "memory");
        }
#else
#pragma unroll
        for (int j = 0; j < 2; ++j) {
            const int flat = j * 256 + tid;
            const int r    = flat >> 5;
            const int col  = (flat & 31) * 4;
            const float4 v = *(const float4*)(base + (size_t)r * NN + col);
            *(float4*)&buf[b][r * KC + col] = v;
        }
#endif
    };

    auto computeChunk = [&](int c, int b) {
        // ---- branchless logsumexp: 8 contiguous cols per thread ----
        {
            const float* rowp = &buf[b][rRow * KC + (tid & 15) * 8];
            const float4 xa = *(const float4*)rowp;
            const float4 xb = *(const float4*)(rowp + 4);
            const float mc =
                fmaxf(fmaxf(fmaxf(xa.x, xa.y), fmaxf(xa.z, xa.w)),
                      fmaxf(fmaxf(xb.x, xb.y), fmaxf(xb.z, xb.w)));
            const float nm = fmaxf(m, mc);
            s = s * __expf(m - nm);               // one rescale per chunk
            s += __expf(xa.x - nm) + __expf(xa.y - nm) +
                 __expf(xa.z - nm) + __expf(xa.w - nm) +
                 __expf(xb.x - nm) + __expf(xb.y - nm) +
                 __expf(xb.z - nm) + __expf(xb.w - nm);
            m = nm;
#if !HAVE_WMMA_F32X4
            const int cb = c * KC + (tid & 15) * 8;
            const float xs[8] = {xa.x, xa.y, xa.z, xa.w, xb.x, xb.y, xb.z, xb.w};
#pragma unroll
            for (int e = 0; e < 8; ++e) {
                const int lk = shLab[cb + e];
                pS += xs[e] * (mat ? shGcl[lk] : 1.f);
                pM += (lk == labR) ? xs[e] : 0.f;
            }
#endif
        }
#if HAVE_WMMA_F32X4
        // ---- masked & (weighted) row sums via V_WMMA_F32_16X16X4_F32 ----
        // A[16x4]: lane(nIdx)=M, VGPR pair = K {kh*2, kh*2+1}
        // B[4x16]: b.x -> K=kh, b.y -> K=2+kh, column = nIdx
#pragma unroll
        for (int q = wave; q < KC / 4; q += 8) {
            const int kl = q * 4;
            const int kg = c * KC + kl;
            v2f a;
            {
                const float2 av =
                    *(const float2*)&buf[b][nIdx * KC + kl + kh * 2];
                a.x = av.x; a.y = av.y;
            }
            const int l0 = shLab[kg + kh];
            const int l1 = shLab[kg + 2 + kh];
            v2f bm, bs;
            bm.x = (l0 == labN) ? 1.f : 0.f;
            bm.y = (l1 == labN) ? 1.f : 0.f;
            if (mat == 0) { bs.x = 1.f;        bs.y = 1.f;        }
            else          { bs.x = shGcl[l0];  bs.y = shGcl[l1];  }
            accM = __builtin_amdgcn_wmma_f32_16x16x4_f32(
                       false, a, false, bm, (short)0, accM, false, false);
            accS = __builtin_amdgcn_wmma_f32_16x16x4_f32(
                       false, a, false, bs, (short)0, accS, false, false);
        }
#endif
    };

#if USE_ASYNC_LDS
    issueChunk(0, 0);
    for (int c = 0; c < NCHUNK; ++c) {
        const int b = c & 1;
        if (c + 1 < NCHUNK) {
            issueChunk(c + 1, b ^ 1);
            asm volatile("s_wait_asynccnt 0x2" ::: "memory");  // chunk c done
        } else {
            asm volatile("s_wait_asynccnt 0x0" ::: "memory");
        }
        __syncthreads();
        computeChunk(c, b);
        __syncthreads();          // buf[b] free before it is re-issued
    }
#else
    for (int c = 0; c < NCHUNK; ++c) {
        issueChunk(c, 0);
        __syncthreads();
        computeChunk(c, 0);
        __syncthreads();
    }
#endif

    // ---- combine (m,s) across the 16 lanes sharing a row ----
#pragma unroll
    for (int off = 8; off; off >>= 1) {
        const float om = __shfl_xor(m, off, 32);
        const float os = __shfl_xor(s, off, 32);
        const float nm = fmaxf(m, om);
        s = s * __expf(m - nm) + os * __expf(om - nm);
        m = nm;
    }
    if ((tid & 15) == 0) shRowL[rRow] = m + logf(s);

#if HAVE_WMMA_F32X4
    // accS: D[m][n] identical over n -> lanes 0 (M=0..7) and 16 (M=8..15)
    if (nIdx == 0) {
#pragma unroll
        for (int i = 0; i < 8; ++i) shPartS[wave * 16 + kh * 8 + i] = accS[i];
    }
    // accM diagonal: lanes 0..7 (M=N=lane, VGPR lane), lanes 24..31
    // (M=N=lane-16, VGPR lane-24)
    {
        const int mrow = (lane < 8) ? lane : ((lane >= 24) ? (lane - 16) : -1);
        if (mrow >= 0) {
            const int idx = mrow & 7;
            shPartM[wave * 16 + mrow] = accM[idx];
        }
    }
#else
#pragma unroll
    for (int off = 8; off; off >>= 1) {
        pS += __shfl_xor(pS, off, 32);
        pM += __shfl_xor(pM, off, 32);
    }
    if ((tid & 15) == 0) {
        shPartS[wave * 16 + rRow] = pS;
        shPartM[wave * 16 + rRow] = pM;
    }
#endif
    __syncthreads();

    // ---- per-row contribution, then fixed-order block sum (deterministic) ----
    if (tid < 16) {
        const int lab = shLab[rowBase + tid];
        float S = 0.f, M = 0.f;
#pragma unroll
        for (int w = 0; w < 8; ++w) {
            S += shPartS[w * 16 + tid];
            M += shPartM[w * 16 + tid];
        }
        const float L = shRowL[tid];
        float contrib;
        if (mat == 0) {
            const float g = gcl[lab], f = fcl[lab];
            contrib = g * S + (f - g) * M - L;          // row term of C1
        } else {
            const float g = gcl[lab];
            contrib = S + E10M1F * g * M - L * tcl[lab]; // row term of C2
        }
        shContrib[tid] = contrib;
    }
    __syncthreads();
    if (tid == 0) {
        float acc = 0.f;
#pragma unroll
        for (int r = 0; r < 16; ++r) acc += shContrib[r];
        blockOut[bid] = acc;
    }
}

// ---------------------------------------------------------------------------
// Kernel 3: deterministic final reduction of 1024 block partials.
// ---------------------------------------------------------------------------
__global__ __launch_bounds__(256)
void kl_finalize(const float* __restrict__ parts, const double* __restrict__ scal,
                 float* __restrict__ out)
{
    __shared__ double red[256];
    const int t = threadIdx.x;
    double s = 0.0;
#pragma unroll
    for (int j = 0; j < 4; ++j) s += (double)parts[t * 4 + j];
    red[t] = s;
    __syncthreads();
    for (int off = 128; off > 0; off >>= 1) {
        if (t < off) red[t] += red[t + off];
        __syncthreads();
    }
    if (t == 0) {
        const double C = red[0];      // C1 + C2
        const double E = scal[0];
        out[0] = (float)((2.0 * E - C) / (2.0 * (double)NN));
    }
}

// ---------------------------------------------------------------------------
extern "C" void kernel_launch(void* const* d_in, const int* in_sizes, int n_in,
                              void* d_out, int out_size, void* d_ws, size_t ws_size,
                              hipStream_t stream)
{
    (void)in_sizes; (void)n_in; (void)out_size; (void)ws_size;
    const float* p1 = (const float*)d_in[0];
    const float* p2 = (const float*)d_in[1];
    const int*   lb = (const int*)d_in[2];

    char* ws = (char*)d_ws;
    float*  gcl   = (float*)(ws + 0);      // 128 floats
    float*  fcl   = (float*)(ws + 512);    // 128 floats
    float*  tcl   = (float*)(ws + 1024);   // 128 floats
    double* scal  = (double*)(ws + 1536);  // E
    float*  parts = (float*)(ws + 1600);   // 1024 block partials

    kl_class_stats<<<1, 256, 0, stream>>>(lb, gcl, fcl, tcl, scal);
    kl_main<<<2 * NBLK_PER_MAT, 256, 0, stream>>>(p1, p2, lb, gcl, fcl, tcl, parts);
    kl_finalize<<<1, 256, 0, stream>>>(parts, scal, (float*)d_out);
}